// GCNLayer_30477087932726
// MI455X (gfx1250) — compile-verified
//
#include <hip/hip_runtime.h>

typedef __attribute__((ext_vector_type(2))) float v2f;
typedef __attribute__((ext_vector_type(8))) float v8f;

#define IN_FT 128
#define OUT_FT 64
#define LDS_STRIDE 132   // 128 + 4 pad: row-to-row bank offset of 4 banks -> no 16-way conflict

// ---------------------------------------------------------------------------
// Kernel 1: h = x @ W + b   using V_WMMA_F32_16X16X4_F32
// Block = 256 threads (8 waves). Each wave owns 16 rows x 64 cols.
// Block covers 128 rows. W transposed into LDS as Wt[n][k].
// A-frag layout (16x4 f32): lane L<16: M=L, v0=K+0, v1=K+1 ; lane L>=16: M=L-16, v0=K+2, v1=K+3
// B-frag layout (4x16 f32): mirrored K split across lane halves, N across lanes 0-15.
// ---------------------------------------------------------------------------
__global__ __launch_bounds__(256) void gcn_gemm_wmma(const float* __restrict__ x,
                                                     const float* __restrict__ W,
                                                     const float* __restrict__ bias,
                                                     float* __restrict__ h,
                                                     int nrows)
{
    __shared__ float Wt[OUT_FT * LDS_STRIDE];

    const int tid = threadIdx.x;

    // Load + transpose W[k][n] -> Wt[n][k]
    for (int i = tid; i < IN_FT * OUT_FT; i += 256) {
        int k = i / OUT_FT;
        int n = i - k * OUT_FT;
        Wt[n * LDS_STRIDE + k] = W[i];
    }
    __syncthreads();

    const int wave = tid >> 5;          // 0..7
    const int lane = tid & 31;
    const int m    = lane & 15;         // row/col within 16
    const int ksel = (lane >> 4) << 1;  // 0 for lanes 0-15, 2 for lanes 16-31
    const int row0 = blockIdx.x * 128 + wave * 16;

    // Clamp A row for loads (stores are guarded; WMMA needs EXEC all-ones)
    int arow = row0 + m;
    if (arow >= nrows) arow = nrows - 1;
    const float* __restrict__ xrow = x + (size_t)arow * IN_FT;

    v8f acc0 = {}, acc1 = {}, acc2 = {}, acc3 = {};

    #pragma unroll 4
    for (int k = 0; k < IN_FT; k += 4) {
        v2f a  = *(const v2f*)(xrow + k + ksel);
        v2f b0 = *(const v2f*)(&Wt[( 0 + m) * LDS_STRIDE + k + ksel]);
        v2f b1 = *(const v2f*)(&Wt[(16 + m) * LDS_STRIDE + k + ksel]);
        v2f b2 = *(const v2f*)(&Wt[(32 + m) * LDS_STRIDE + k + ksel]);
        v2f b3 = *(const v2f*)(&Wt[(48 + m) * LDS_STRIDE + k + ksel]);
        acc0 = __builtin_amdgcn_wmma_f32_16x16x4_f32(false, a, false, b0, (short)0, acc0, false, false);
        acc1 = __builtin_amdgcn_wmma_f32_16x16x4_f32(false, a, false, b1, (short)0, acc1, false, false);
        acc2 = __builtin_amdgcn_wmma_f32_16x16x4_f32(false, a, false, b2, (short)0, acc2, false, false);
        acc3 = __builtin_amdgcn_wmma_f32_16x16x4_f32(false, a, false, b3, (short)0, acc3, false, false);
    }

    // C/D layout: VGPR r -> (M=r, N=lane) lanes 0-15 ; (M=r+8, N=lane-16) lanes 16-31
    const float bv0 = bias[ 0 + m];
    const float bv1 = bias[16 + m];
    const float bv2 = bias[32 + m];
    const float bv3 = bias[48 + m];
    const int rbase = row0 + ((lane >> 4) << 3);

    #pragma unroll
    for (int r = 0; r < 8; ++r) {
        int row = rbase + r;
        if (row < nrows) {
            float* __restrict__ hrow = h + (size_t)row * OUT_FT;
            hrow[ 0 + m] = acc0[r] + bv0;
            hrow[16 + m] = acc1[r] + bv1;
            hrow[32 + m] = acc2[r] + bv2;
            hrow[48 + m] = acc3[r] + bv3;
        }
    }
}

// ---------------------------------------------------------------------------
// Kernel 0: zero the aggregation buffer (d_out)
// ---------------------------------------------------------------------------
__global__ __launch_bounds__(256) void gcn_zero(float4* __restrict__ out, int n4)
{
    int i = blockIdx.x * 256 + threadIdx.x;
    if (i < n4) out[i] = make_float4(0.f, 0.f, 0.f, 0.f);
}

// ---------------------------------------------------------------------------
// Kernel 2: out[row[e]] += val[e] * h[col[e]]   (one feature per lane)
// h (25.6 MB) fits in the 192 MB L2 -> gather served from L2.
// ---------------------------------------------------------------------------
__global__ __launch_bounds__(256) void gcn_spmm_scatter(const int* __restrict__ erow,
                                                        const int* __restrict__ ecol,
                                                        const float* __restrict__ eval,
                                                        const float* __restrict__ h,
                                                        float* __restrict__ out,
                                                        int E)
{
    int gid = blockIdx.x * 256 + threadIdx.x;
    int e = gid >> 6;          // edge index
    if (e >= E) return;
    int f = gid & 63;          // feature index (coalesced across wave)
    int r = erow[e];
    int c = ecol[e];
    float v = eval[e];
    float msg = v * h[(size_t)c * OUT_FT + f];
    atomicAdd(&out[(size_t)r * OUT_FT + f], msg);
}

// ---------------------------------------------------------------------------
// Kernel 3: in-place PReLU with single shared slope
// ---------------------------------------------------------------------------
__global__ __launch_bounds__(256) void gcn_prelu(float* __restrict__ out,
                                                 const float* __restrict__ alpha,
                                                 int n)
{
    int i = blockIdx.x * 256 + threadIdx.x;
    if (i < n) {
        float a = alpha[0];
        float v = out[i];
        out[i] = (v >= 0.f) ? v : a * v;
    }
}

// ---------------------------------------------------------------------------
extern "C" void kernel_launch(void* const* d_in, const int* in_sizes, int n_in,
                              void* d_out, int out_size, void* d_ws, size_t ws_size,
                              hipStream_t stream)
{
    const float* x     = (const float*)d_in[0];
    const float* W     = (const float*)d_in[1];
    const float* bias  = (const float*)d_in[2];
    const float* alpha = (const float*)d_in[3];
    const int*   erow  = (const int*)  d_in[4];
    const int*   ecol  = (const int*)  d_in[5];
    const float* eval  = (const float*)d_in[6];

    const int N = in_sizes[0] / IN_FT;   // 100000
    const int E = in_sizes[4];           // 1600000

    float* out = (float*)d_out;          // [N, OUT_FT]
    float* h   = (float*)d_ws;           // [N, OUT_FT] scratch (25.6 MB)

    // 0) zero aggregation target
    int n4 = out_size / 4;
    gcn_zero<<<(n4 + 255) / 256, 256, 0, stream>>>((float4*)out, n4);

    // 1) h = x @ W + b   (WMMA)
    gcn_gemm_wmma<<<(N + 127) / 128, 256, 0, stream>>>(x, W, bias, h, N);

    // 2) scatter-add SpMM
    long total = (long)E * OUT_FT;
    gcn_spmm_scatter<<<(unsigned)((total + 255) / 256), 256, 0, stream>>>(erow, ecol, eval, h, out, E);

    // 3) PReLU in place
    gcn_prelu<<<(out_size + 255) / 256, 256, 0, stream>>>(out, alpha, out_size);
}